// LorentzianGATLayer_51857435131908
// MI455X (gfx1250) — compile-verified
//
#include <hip/hip_runtime.h>
#include <hip/hip_bf16.h>

// Problem constants from the reference
#define BD    4
#define NNODE 16384
#define DD    128
#define NSEG  (BD * NNODE)       // 65536 rows
#define NEDGE 1048576

typedef __attribute__((ext_vector_type(2))) float v2f;
typedef __attribute__((ext_vector_type(8))) float v8f;

// Order-preserving float -> int key so we can use integer atomicMax for the
// per-destination segment max (scores can be negative).
__device__ __forceinline__ int fkey(float f) {
  int i = __float_as_int(f);
  return i >= 0 ? i : (i ^ 0x7fffffff);
}
__device__ __forceinline__ float fdecode(int k) {
  return __int_as_float(k >= 0 ? k : (k ^ 0x7fffffff));
}

// ---------------------------------------------------------------------------
// 0) init workspace: agg = 0, mkey = INT_MIN, denom = 0 (every launch)
// ---------------------------------------------------------------------------
__global__ void __launch_bounds__(256)
init_ws_kernel(float* __restrict__ agg, int* __restrict__ mkey,
               float* __restrict__ denom) {
  int i = blockIdx.x * 256 + threadIdx.x;
  if (i < NSEG * DD) agg[i] = 0.0f;
  if (i < NSEG) { mkey[i] = (int)0x80000000; denom[i] = 0.0f; }
}

// ---------------------------------------------------------------------------
// 1) h = X @ Wt + bt   (M x 128 @ 128 x 128, fp32 WMMA 16x16x4)
//    block = 256 thr = 8 waves; block -> 16-row strip; wave w -> cols [16w,16w+16)
// ---------------------------------------------------------------------------
__global__ void __launch_bounds__(256)
gemm_h_kernel(const float* __restrict__ X, const float* __restrict__ W,
              const float* __restrict__ bias, float* __restrict__ Y) {
  const int lane = threadIdx.x & 31;
  const int wave = threadIdx.x >> 5;
  const int row0 = blockIdx.x << 4;
  const int col0 = wave << 4;
  const int m  = lane & 15;          // row (A) / col (B,C,D) within tile
  const int kh = (lane >> 4) << 1;   // K sub-pair: lanes 0-15 -> K0/K1, 16-31 -> K2/K3
  const float* xr = X + (size_t)(row0 + m) * DD;

  v8f acc = {};
#pragma unroll
  for (int k = 0; k < DD; k += 4) {
    v2f a = *(const v2f*)(xr + k + kh);               // A: 16x4 fragment
    v2f b;                                            // B: 4x16 fragment
    b.x = W[(size_t)(k + kh    ) * DD + col0 + m];
    b.y = W[(size_t)(k + kh + 1) * DD + col0 + m];
    acc = __builtin_amdgcn_wmma_f32_16x16x4_f32(false, a, false, b,
                                                (short)0, acc, false, false);
  }

  const int   rb = (lane >> 4) << 3;   // lanes 16-31 hold rows M+8
  const float bv = bias[col0 + m];
#pragma unroll
  for (int r = 0; r < 8; ++r)
    Y[(size_t)(row0 + rb + r) * DD + col0 + m] = acc[r] + bv;
}

// ---------------------------------------------------------------------------
// 2) per-edge Lorentzian score + segment max (8 lanes per edge, wave32 shuffles)
// ---------------------------------------------------------------------------
__global__ void __launch_bounds__(256)
edge_score_kernel(const float* __restrict__ h, const int* __restrict__ adj,
                  const float* __restrict__ vals, float* __restrict__ score,
                  int* __restrict__ mkey, int E) {
  int t   = blockIdx.x * 256 + threadIdx.x;
  int e   = t >> 3;
  int sub = t & 7;
  if (e >= E) return;
  const int b  = adj[3 * e + 0];
  const int s  = adj[3 * e + 1];
  const int dd = adj[3 * e + 2];
  const size_t sg = (size_t)b * NNODE + s;
  const size_t dg = (size_t)b * NNODE + dd;
  const float4* hs = (const float4*)(h + sg * DD) + sub * 4;  // 16 dims/lane
  const float4* hd = (const float4*)(h + dg * DD) + sub * 4;

  float acc = 0.0f;
#pragma unroll
  for (int i = 0; i < 4; ++i) {
    float4 a = hs[i], c = hd[i];
    acc += a.x * c.x + a.y * c.y + a.z * c.z + a.w * c.w;
  }
  if (sub == 0) {                       // Lorentzian signature: -2*h0*h0'
    float4 a0 = hs[0], c0 = hd[0];
    acc -= 2.0f * a0.x * c0.x;
  }
  // reduce the 8-lane group (wave32)
  acc += __shfl_xor(acc, 1, 32);
  acc += __shfl_xor(acc, 2, 32);
  acc += __shfl_xor(acc, 4, 32);
  if (sub == 0) {
    float sc = acc * vals[e];
    score[e] = sc;
    atomicMax(mkey + dg, fkey(sc));
  }
}

// ---------------------------------------------------------------------------
// 3) e = exp(score - max[dst]); denom[dst] += e; score buffer now holds e
// ---------------------------------------------------------------------------
__global__ void __launch_bounds__(256)
edge_exp_kernel(const int* __restrict__ adj, float* __restrict__ score,
                const int* __restrict__ mkey, float* __restrict__ denom, int E) {
  int e = blockIdx.x * 256 + threadIdx.x;
  if (e >= E) return;
  const int b  = adj[3 * e + 0];
  const int dd = adj[3 * e + 2];
  const size_t dg = (size_t)b * NNODE + dd;
  const float mf = fdecode(mkey[dg]);
  const float ev = __expf(score[e] - mf);
  score[e] = ev;
  atomicAdd(denom + dg, ev);
}

// ---------------------------------------------------------------------------
// 4) agg[dst] += (e/denom[dst]) * h[src]   (L2-resident float atomics)
// ---------------------------------------------------------------------------
__global__ void __launch_bounds__(256)
edge_agg_kernel(const float* __restrict__ h, const int* __restrict__ adj,
                const float* __restrict__ score, const float* __restrict__ denom,
                float* __restrict__ agg, int E) {
  int t   = blockIdx.x * 256 + threadIdx.x;
  int e   = t >> 3;
  int sub = t & 7;
  if (e >= E) return;
  const int b  = adj[3 * e + 0];
  const int s  = adj[3 * e + 1];
  const int dd = adj[3 * e + 2];
  const size_t sg = (size_t)b * NNODE + s;
  const size_t dg = (size_t)b * NNODE + dd;
  const float coef = score[e] / denom[dg];
  const float4* hs = (const float4*)(h + sg * DD) + sub * 4;
  float* ag = agg + dg * DD + sub * 16;
#pragma unroll
  for (int i = 0; i < 4; ++i) {
    float4 a = hs[i];
    atomicAdd(ag + 4 * i + 0, coef * a.x);
    atomicAdd(ag + 4 * i + 1, coef * a.y);
    atomicAdd(ag + 4 * i + 2, coef * a.z);
    atomicAdd(ag + 4 * i + 3, coef * a.w);
  }
}

// ---------------------------------------------------------------------------
// 5) fused: act = relu((gate*agg) @ Wa + ba) [staged in LDS]; out = act @ Wo + bo
// ---------------------------------------------------------------------------
__global__ void __launch_bounds__(256)
out_fused_kernel(const float* __restrict__ agg, const float* __restrict__ gate,
                 const float* __restrict__ Wa, const float* __restrict__ ba,
                 const float* __restrict__ Wo, const float* __restrict__ bo,
                 float* __restrict__ out) {
  __shared__ float act[16][DD];   // 8 KB activation strip
  const int lane = threadIdx.x & 31;
  const int wave = threadIdx.x >> 5;
  const int row0 = blockIdx.x << 4;
  const int col0 = wave << 4;
  const int m  = lane & 15;
  const int kh = (lane >> 4) << 1;
  const int rb = (lane >> 4) << 3;

  // GEMM 1: (gate*agg) @ Wa
  const float g  = gate[row0 + m];
  const float* ar = agg + (size_t)(row0 + m) * DD;
  v8f acc = {};
#pragma unroll
  for (int k = 0; k < DD; k += 4) {
    v2f a = *(const v2f*)(ar + k + kh);
    a.x *= g; a.y *= g;
    v2f b;
    b.x = Wa[(size_t)(k + kh    ) * DD + col0 + m];
    b.y = Wa[(size_t)(k + kh + 1) * DD + col0 + m];
    acc = __builtin_amdgcn_wmma_f32_16x16x4_f32(false, a, false, b,
                                                (short)0, acc, false, false);
  }
  const float bav = ba[col0 + m];
#pragma unroll
  for (int r = 0; r < 8; ++r) {
    float v = acc[r] + bav;
    act[rb + r][col0 + m] = v > 0.0f ? v : 0.0f;   // ReLU into LDS
  }
  __syncthreads();

  // GEMM 2: act @ Wo (A fragments from LDS)
  v8f acc2 = {};
#pragma unroll
  for (int k = 0; k < DD; k += 4) {
    v2f a;
    a.x = act[m][k + kh];
    a.y = act[m][k + kh + 1];
    v2f b;
    b.x = Wo[(size_t)(k + kh    ) * DD + col0 + m];
    b.y = Wo[(size_t)(k + kh + 1) * DD + col0 + m];
    acc2 = __builtin_amdgcn_wmma_f32_16x16x4_f32(false, a, false, b,
                                                 (short)0, acc2, false, false);
  }
  const float bov = bo[col0 + m];
#pragma unroll
  for (int r = 0; r < 8; ++r)
    out[(size_t)(row0 + rb + r) * DD + col0 + m] = acc2[r] + bov;
}

// ---------------------------------------------------------------------------
extern "C" void kernel_launch(void* const* d_in, const int* in_sizes, int n_in,
                              void* d_out, int out_size, void* d_ws, size_t ws_size,
                              hipStream_t stream) {
  const float* X    = (const float*)d_in[0];   // node_features (B,N,D)
  const int*   adj  = (const int*)  d_in[1];   // adj_indices (E,3) int32
  const float* vals = (const float*)d_in[2];   // adj_values (E,)
  // d_in[3] = adj_dense_shape (unused)
  const float* attw = (const float*)d_in[4];   // attention_weights (B,N,1): gate == itself
  const float* Wt   = (const float*)d_in[5];
  const float* bt   = (const float*)d_in[6];
  const float* Wa   = (const float*)d_in[7];
  const float* ba   = (const float*)d_in[8];
  const float* Wo   = (const float*)d_in[9];
  const float* bo   = (const float*)d_in[10];
  float* out = (float*)d_out;

  const int E = in_sizes[2];            // 1048576
  const int M = in_sizes[0] / DD;       // 65536 rows

  // workspace layout (~68.5 MB)
  char*  ws    = (char*)d_ws;
  float* h     = (float*)ws;                                   // 32 MB
  float* agg   = (float*)(ws + (size_t)NSEG * DD * 4);         // 32 MB
  float* score = (float*)(ws + 2 * (size_t)NSEG * DD * 4);     //  4 MB
  int*   mkey  = (int*)  (ws + 2 * (size_t)NSEG * DD * 4 + (size_t)NEDGE * 4);
  float* denom = (float*)(ws + 2 * (size_t)NSEG * DD * 4 + (size_t)NEDGE * 4
                             + (size_t)NSEG * 4);

  const int initBlocks  = (NSEG * DD + 255) / 256;     // 32768
  const int gemmBlocks  = M / 16;                      // 4096 strips
  const int scoreBlocks = (E * 8 + 255) / 256;         // 32768
  const int expBlocks   = (E + 255) / 256;             // 4096

  init_ws_kernel  <<<initBlocks, 256, 0, stream>>>(agg, mkey, denom);
  gemm_h_kernel   <<<gemmBlocks, 256, 0, stream>>>(X, Wt, bt, h);
  edge_score_kernel<<<scoreBlocks, 256, 0, stream>>>(h, adj, vals, score, mkey, E);
  edge_exp_kernel <<<expBlocks, 256, 0, stream>>>(adj, score, mkey, denom, E);
  edge_agg_kernel <<<scoreBlocks, 256, 0, stream>>>(h, adj, score, denom, agg, E);
  out_fused_kernel<<<gemmBlocks, 256, 0, stream>>>(agg, attw, Wa, ba, Wo, bo, out);
}